// MemKDMClassModel_74285754352156
// MI455X (gfx1250) — compile-verified
//
#include <hip/hip_runtime.h>

// Problem constants (from reference)
#define N_COMP 128
#define ENC    512
#define DIM_Y  100
#define EPS    1e-10f

typedef __attribute__((ext_vector_type(2))) float v2f;
typedef __attribute__((ext_vector_type(8))) float v8f;

// One block per sample b. 256 threads = 8 waves (wave32); each wave owns 16
// neighbors and runs the K-loop with V_WMMA_F32_16X16X4_F32.
__launch_bounds__(256, 4)
__global__ void kdm_probs_kernel(const float* __restrict__ x_enc,
                                 const float* __restrict__ x_neigh,
                                 const float* __restrict__ sigma,
                                 const int*   __restrict__ y_neigh,
                                 float* __restrict__ out)
{
    __shared__ float lds_xenc[ENC];     // staged x_enc[b]
    __shared__ float lds_xn[N_COMP];    // per-neighbor dot(x, neigh)
    __shared__ float lds_n2[N_COMP];    // per-neighbor ||neigh||^2
    __shared__ float lds_w[N_COMP];     // unnormalized weights
    __shared__ float lds_probs[DIM_Y];  // class accumulator
    __shared__ float lds_x2;            // ||x||^2
    __shared__ float lds_wsum;          // sum of weights

    const int b    = blockIdx.x;
    const int tid  = threadIdx.x;
    const int lane = tid & 31;
    const int wave = tid >> 5;

    // ---- init LDS accumulators ----
    if (tid == 0) { lds_x2 = 0.0f; lds_wsum = 0.0f; }
    if (tid < DIM_Y) lds_probs[tid] = 0.0f;
    __syncthreads();

    // ---- stage x_enc[b] into LDS, accumulate ||x||^2 ----
    const float* xe = x_enc + (size_t)b * ENC;
    float v0 = xe[tid];
    float v1 = xe[tid + 256];
    lds_xenc[tid]       = v0;
    lds_xenc[tid + 256] = v1;
    float px2 = fmaf(v0, v0, v1 * v1);
    #pragma unroll
    for (int off = 16; off > 0; off >>= 1)
        px2 += __shfl_down(px2, off, 32);
    if (lane == 0) atomicAdd(&lds_x2, px2);   // ds_add_f32
    __syncthreads();                          // x_enc visible to all waves

    // ---- main K-loop: WMMA f32 16x16x4, A = 16 neighbor rows, B = x_enc
    //      broadcast into all 16 columns. n2 accumulated on the A registers. ----
    const int  nbase = wave * 16;             // this wave's neighbor tile
    const int  row   = nbase + (lane & 15);
    const bool hi    = (lane >= 16);
    // A layout (16x4 f32): lanes 0-15 hold K={k,k+1}, lanes 16-31 hold K={k+2,k+3}
    const float* arow = x_neigh + ((size_t)b * N_COMP + row) * ENC + (hi ? 2 : 0);
    const float* brow = lds_xenc + (hi ? 2 : 0);

    v8f   acc = {};
    float pn2 = 0.0f;

    #pragma unroll 8
    for (int k = 0; k < ENC; k += 4) {
        v2f a  = *(const v2f*)(arow + k);     // global_load_b64 (streams row)
        v2f bb = *(const v2f*)(brow + k);     // ds_load_b64 (broadcast)
        pn2 = fmaf(a.x, a.x, pn2);
        pn2 = fmaf(a.y, a.y, pn2);
        // D = A * B + C   (v_wmma_f32_16x16x4_f32)
        acc = __builtin_amdgcn_wmma_f32_16x16x4_f32(
                  /*neg_a=*/false, a, /*neg_b=*/false, bb,
                  /*c_mod=*/(short)0, acc, /*reuse_a=*/false, /*reuse_b=*/false);
    }

    // ---- n2: combine half-wave partials (lane L gets lane L+16's partial) ----
    pn2 += __shfl_down(pn2, 16, 32);
    if (lane < 16) lds_n2[nbase + lane] = pn2;

    // ---- xn: all B columns identical, so column 0 carries valid results.
    //      lane 0 holds rows 0..7 in acc[0..7], lane 16 holds rows 8..15. ----
    if (lane == 0) {
        #pragma unroll
        for (int r = 0; r < 8; ++r) lds_xn[nbase + r] = acc[r];
    } else if (lane == 16) {
        #pragma unroll
        for (int r = 0; r < 8; ++r) lds_xn[nbase + 8 + r] = acc[r];
    }
    __syncthreads();

    // ---- epilogue: RBF kernel, squared-kernel weights, normalization ----
    const float s      = sigma[0];
    const float invden = 1.0f / (2.0f * s * s);
    if (tid < N_COMP) {
        float d2 = lds_x2 + lds_n2[tid] - 2.0f * lds_xn[tid];
        d2 = fmaxf(d2, 0.0f);
        float kk = __expf(-d2 * invden);
        float w  = kk * kk;
        lds_w[tid] = w;
        float ws = w;
        #pragma unroll
        for (int off = 16; off > 0; off >>= 1)
            ws += __shfl_down(ws, off, 32);
        if (lane == 0) atomicAdd(&lds_wsum, ws);
    }
    __syncthreads();

    // ---- scatter normalized weights into class bins (LDS float atomics) ----
    if (tid < N_COMP) {
        float wn = lds_w[tid] / (lds_wsum + EPS);
        int   y  = y_neigh[(size_t)b * N_COMP + tid];
        atomicAdd(&lds_probs[y], wn);
    }
    __syncthreads();

    // ---- coalesced output: 100 floats per sample ----
    if (tid < DIM_Y) out[(size_t)b * DIM_Y + tid] = lds_probs[tid];
}

extern "C" void kernel_launch(void* const* d_in, const int* in_sizes, int n_in,
                              void* d_out, int out_size, void* d_ws, size_t ws_size,
                              hipStream_t stream) {
    const float* x_enc   = (const float*)d_in[0];  // (BS, ENC)
    const float* x_neigh = (const float*)d_in[1];  // (BS, N_COMP, ENC)
    const float* sigma   = (const float*)d_in[2];  // (1,)
    const int*   y_neigh = (const int*)d_in[3];    // (BS, N_COMP) int32
    float*       out     = (float*)d_out;          // (BS, DIM_Y)

    const int bs = in_sizes[0] / ENC;              // 4096
    kdm_probs_kernel<<<bs, 256, 0, stream>>>(x_enc, x_neigh, sigma, y_neigh, out);
}